// SpatialEngram1D_38199439131348
// MI455X (gfx1250) — compile-verified
//
#include <hip/hip_runtime.h>

typedef __attribute__((ext_vector_type(2))) float v2f;
typedef __attribute__((ext_vector_type(8))) float v8f;

#define B_    32
#define C_    64
#define L_    16384
#define E_    8
#define NPAT  1000

// ---------------------------------------------------------------------------
// Kernel 1: ch_sum[b][l] = sum_c x[b][c][l].  float4 along L, coalesced reads.
// Reads 128 MB, writes 2 MB -> pure bandwidth.
// ---------------------------------------------------------------------------
__global__ void engram_chsum_kernel(const float* __restrict__ x,
                                    float* __restrict__ chsum) {
    const int t  = blockIdx.x * blockDim.x + threadIdx.x;   // 0 .. B*L/4-1
    const int b  = t / (L_ / 4);
    const int l4 = t % (L_ / 4);
    const float4* xp = (const float4*)(x + (size_t)b * C_ * L_) + l4;
    float4 acc = make_float4(0.f, 0.f, 0.f, 0.f);
#pragma unroll 8
    for (int c = 0; c < C_; ++c) {
        float4 v = xp[(size_t)c * (L_ / 4)];
        acc.x += v.x; acc.y += v.y; acc.z += v.z; acc.w += v.w;
    }
    ((float4*)(chsum + (size_t)b * L_))[l4] = acc;
}

// ---------------------------------------------------------------------------
// Kernel 2: one wave32 per 16 sequence positions.
//   win_sum -> hash idx -> gather emb row (B frag, K=8 split into 2x K=4)
//   D[ch][l] = proj_w(16x8) x emb^T(8x16) + bias, 4 channel tiles per wave.
//   fp32 WMMA 16x16x4 chained twice => exact-precision projection.
//   D layout: N(=seq) across lanes -> per-VGPR stores are coalesced 64B runs.
// ---------------------------------------------------------------------------
__global__ void engram_proj_wmma_kernel(const float* __restrict__ chsum,
                                        const float* __restrict__ emb,   // [1000][8]
                                        const float* __restrict__ pw,    // [64][8]
                                        const float* __restrict__ pb,    // [64]
                                        float* __restrict__ out) {
    const int lane = threadIdx.x & 31;
    const int wave = (blockIdx.x * blockDim.x + threadIdx.x) >> 5;
    const int tilesPerBatch = L_ / 16;                 // 1024
    const int b  = wave / tilesPerBatch;
    const int l0 = (wave % tilesPerBatch) * 16;

    const int n = lane & 15;   // seq position within tile (B/D cols), ch row for A
    const int h = lane >> 4;   // lane half selects K pairs / row offset +8

    // ---- window sum + hash for seq position l0+n (both halves compute it) ----
    const float* cs = chsum + (size_t)b * L_;
    const int l = l0 + n;
    float wsum = 0.f;
#pragma unroll
    for (int w = -2; w <= 1; ++w) {
        int p = l + w;
        p = p < 0 ? 0 : (p > L_ - 1 ? L_ - 1 : p);
        wsum += cs[p];
    }
    // int64 trunc toward zero, then nonnegative mod (torch semantics)
    long long hv = (long long)(wsum * 31.0f);
    int idx = (int)(hv % NPAT);
    if (idx < 0) idx += NPAT;

    // ---- B fragment: emb row for this seq position, 4x16 per K-chunk ----
    // B layout (32-bit 4x16): VGPR0 = K{0|2}, VGPR1 = K{1|3} by lane half.
    const float* er = emb + (size_t)idx * E_;
    v2f b_lo, b_hi;
    b_lo.x = er[2 * h];     b_lo.y = er[2 * h + 1];
    b_hi.x = er[4 + 2 * h]; b_hi.y = er[4 + 2 * h + 1];

    float* o = out + (size_t)b * C_ * L_ + l0;

#pragma unroll
    for (int ct = 0; ct < 4; ++ct) {
        const int ch0 = ct * 16;

        // A fragment (16x4, MxK): lanes 0-15 hold K0/K1, lanes 16-31 K2/K3.
        const float* wr = pw + (size_t)(ch0 + n) * E_;
        v2f a_lo, a_hi;
        a_lo.x = wr[2 * h];     a_lo.y = wr[2 * h + 1];
        a_hi.x = wr[4 + 2 * h]; a_hi.y = wr[4 + 2 * h + 1];

        // Seed accumulator with bias: row M = r + 8h (channel).
        v8f c;
#pragma unroll
        for (int r = 0; r < 8; ++r) c[r] = pb[ch0 + r + 8 * h];

        // D = A x B + C, K=8 via two chained K=4 fp32 WMMAs.
        c = __builtin_amdgcn_wmma_f32_16x16x4_f32(false, a_lo, false, b_lo,
                                                  (short)0, c, false, false);
        c = __builtin_amdgcn_wmma_f32_16x16x4_f32(false, a_hi, false, b_hi,
                                                  (short)0, c, false, false);

        // Store: c[r] -> out[b, ch0 + r + 8h, l0 + n]; 16 lanes = 64B run.
#pragma unroll
        for (int r = 0; r < 8; ++r)
            o[(size_t)(ch0 + r + 8 * h) * L_ + n] = c[r];
    }
}

// ---------------------------------------------------------------------------
extern "C" void kernel_launch(void* const* d_in, const int* in_sizes, int n_in,
                              void* d_out, int out_size, void* d_ws, size_t ws_size,
                              hipStream_t stream) {
    (void)in_sizes; (void)n_in; (void)out_size; (void)ws_size;
    const float* x   = (const float*)d_in[0];
    const float* emb = (const float*)d_in[1];
    const float* pw  = (const float*)d_in[2];
    const float* pb  = (const float*)d_in[3];
    float* out   = (float*)d_out;
    float* chsum = (float*)d_ws;                       // needs B*L*4 = 2 MB

    const int t1 = B_ * L_ / 4;                        // 131072 threads
    engram_chsum_kernel<<<t1 / 256, 256, 0, stream>>>(x, chsum);

    const int waves = B_ * (L_ / 16);                  // 32768 waves
    engram_proj_wmma_kernel<<<waves * 32 / 256, 256, 0, stream>>>(
        chsum, emb, pw, pb, out);
}